// GraphAtt_5609227288951
// MI455X (gfx1250) — compile-verified
//
#include <hip/hip_runtime.h>
#include <hip/hip_bf16.h>

typedef __attribute__((ext_vector_type(2))) float v2f;
typedef __attribute__((ext_vector_type(8))) float v8f;

#define C_DIM 128   // C_IN == C_OUT == 128

// ---------------------------------------------------------------------------
// Kernel 1: proj = leaky_relu(word_vec @ W^T + b)   via V_WMMA_F32_16X16X4_F32
// One wave computes a 16x16 output tile; 8 waves/block cover 128 output cols
// for a 16-row stripe. Writes result to BOTH d_out and a scratch snapshot
// (pre-scatter values needed by the aggregation kernel).
// ---------------------------------------------------------------------------
__global__ __launch_bounds__(256) void graphatt_proj_kernel(
    const float* __restrict__ word_vec,   // [N, 128]
    const float* __restrict__ W,          // [128, 128] row-major (C_OUT, C_IN)
    const float* __restrict__ bias,       // [128]
    float* __restrict__ out,              // [N, 128]
    float* __restrict__ proj_copy,        // [N, 128] snapshot in ws
    int N)
{
    const int lane = threadIdx.x & 31;
    const int wave = threadIdx.x >> 5;        // 0..7 -> output-col tile
    const int row0 = blockIdx.x * 16;
    const int o0   = wave * 16;
    const int hi   = lane >> 4;               // 0: lanes 0-15, 1: lanes 16-31
    const int lr   = lane & 15;

    if (row0 + 16 > N) return;                // N is a multiple of 16 anyway

    // A fragment source: word_vec[row0+lr][k0 + 2*hi .. +1]
    const float* Aptr = word_vec + (size_t)(row0 + lr) * C_DIM + 2 * hi;
    // B fragment source: B[k][n] = W[o0+n][k]  (out = A @ W^T)
    const float* Bptr = W        + (size_t)(o0   + lr) * C_DIM + 2 * hi;

    v8f c = {0.f, 0.f, 0.f, 0.f, 0.f, 0.f, 0.f, 0.f};

    #pragma unroll
    for (int k0 = 0; k0 < C_DIM; k0 += 4) {
        v2f a = *(const v2f*)(Aptr + k0);
        v2f b = *(const v2f*)(Bptr + k0);
        // D = A(16x4) x B(4x16) + C, fp32 accumulate
        c = __builtin_amdgcn_wmma_f32_16x16x4_f32(
                /*neg_a=*/false, a, /*neg_b=*/false, b,
                /*c_mod=*/(short)0, c, /*reuse_a=*/false, /*reuse_b=*/false);
    }

    // D layout: VGPR j -> row (j + 8*hi), col lr
    const float bv = bias[o0 + lr];
    #pragma unroll
    for (int j = 0; j < 8; ++j) {
        float v = c[j] + bv;
        v = (v > 0.f) ? v : 0.2f * v;         // LeakyReLU(0.2)
        const size_t idx = (size_t)(row0 + j + 8 * hi) * C_DIM + (o0 + lr);
        out[idx]       = v;
        proj_copy[idx] = v;
    }
}

// ---------------------------------------------------------------------------
// Kernel 2: attention scores -> masked softmax -> weighted aggregation ->
// scatter into d_out at src rows. One wave32 per source s.
//   Phase 1: lane = neighbor index (32 neighbors), dot(q, k) over 128 ch.
//   Phase 2: lane = 4-channel chunk, accumulate prob-weighted proj rows
//            (read from the immutable snapshot -> matches pre-scatter ref).
// ---------------------------------------------------------------------------
__global__ __launch_bounds__(256) void graphatt_attn_kernel(
    const float* __restrict__ word_vec,    // [N, 128]
    const int*   __restrict__ src_idx,     // [S]
    const int*   __restrict__ neighs_idx,  // [S, 32]
    const int*   __restrict__ src_mask,    // [S, 32]
    const float* __restrict__ proj,        // [N, 128] snapshot
    float* __restrict__ out,               // [N, 128]
    int S)
{
    const int lane = threadIdx.x & 31;
    const int wave = threadIdx.x >> 5;
    const int s = blockIdx.x * 8 + wave;
    if (s >= S) return;

    const int my_src   = src_idx[s];
    const int my_neigh = neighs_idx[s * 32 + lane];

    // ---- Phase 1: score[lane] = 5 * dot(q, k_lane), masked ----
    const float* qrow = word_vec + (size_t)my_src   * C_DIM;  // uniform addr -> bcast
    const float* krow = word_vec + (size_t)my_neigh * C_DIM;  // L2-resident gather
    float acc = 0.f;
    #pragma unroll
    for (int cidx = 0; cidx < C_DIM; cidx += 4) {
        const float4 qv = *(const float4*)(qrow + cidx);
        const float4 kv = *(const float4*)(krow + cidx);
        acc += qv.x * kv.x + qv.y * kv.y + qv.z * kv.z + qv.w * kv.w;
    }
    const float score = (src_mask[s * 32 + lane] == 1) ? acc * 5.0f : -1.0e6f;

    // ---- wave32 softmax over the 32 neighbor scores ----
    float m = score;
    #pragma unroll
    for (int off = 16; off > 0; off >>= 1) m = fmaxf(m, __shfl_xor(m, off, 32));
    const float e = expf(score - m);
    float sum = e;
    #pragma unroll
    for (int off = 16; off > 0; off >>= 1) sum += __shfl_xor(sum, off, 32);
    const float prob = e / sum;

    // ---- Phase 2: agg[c] = sum_k prob_k * proj[neigh_k][c] ----
    // lane owns channels [4*lane, 4*lane+4): wave covers all 128, coalesced.
    float4 aggv = make_float4(0.f, 0.f, 0.f, 0.f);
    #pragma unroll
    for (int k = 0; k < 32; ++k) {
        const float p = __shfl(prob,     k, 32);
        const int   r = __shfl(my_neigh, k, 32);
        const float4 pv = *(const float4*)(proj + (size_t)r * C_DIM + lane * 4);
        aggv.x += p * pv.x;
        aggv.y += p * pv.y;
        aggv.z += p * pv.z;
        aggv.w += p * pv.w;
    }

    // scatter: src indices are unique -> no write conflicts
    *(float4*)(out + (size_t)my_src * C_DIM + lane * 4) = aggv;
}

// ---------------------------------------------------------------------------
extern "C" void kernel_launch(void* const* d_in, const int* in_sizes, int n_in,
                              void* d_out, int out_size, void* d_ws, size_t ws_size,
                              hipStream_t stream) {
    const float* word_vec   = (const float*)d_in[0];   // [N, 128] f32
    const int*   src_idx    = (const int*)  d_in[1];   // [S]
    const int*   neighs_idx = (const int*)  d_in[2];   // [S, 32]
    const int*   src_mask   = (const int*)  d_in[3];   // [S, 32]
    const float* W_before   = (const float*)d_in[4];   // [128, 128]
    const float* b_before   = (const float*)d_in[5];   // [128]

    float* out       = (float*)d_out;
    float* proj_copy = (float*)d_ws;                   // N*128*4 = 51.2 MB snapshot

    const int N = in_sizes[0] / C_DIM;                 // 100000
    const int S = in_sizes[1];                         // 50000

    // Kernel 1: projection GEMM (WMMA f32), 16-row stripes, 8 waves/block.
    const int grid1 = (N + 15) / 16;                   // 6250
    graphatt_proj_kernel<<<grid1, 256, 0, stream>>>(
        word_vec, W_before, b_before, out, proj_copy, N);

    // Kernel 2: attention + softmax + aggregation + scatter, 1 wave per source.
    const int grid2 = (S + 7) / 8;                     // 6250
    graphatt_attn_kernel<<<grid2, 256, 0, stream>>>(
        word_vec, src_idx, neighs_idx, src_mask, proj_copy, out, S);
}